// STGI_59167469470005
// MI455X (gfx1250) — compile-verified
//
#include <hip/hip_runtime.h>
#include <math.h>

typedef __attribute__((ext_vector_type(2))) float v2f;
typedef __attribute__((ext_vector_type(8))) float v8f;

#define T_STEPS 12
#define N_NODES 10000
#define F_DIM   128

// ---------------------------------------------------------------------------
// degree: deg[dst] += ew  (gcn_norm, add_self_loops=False)
// ---------------------------------------------------------------------------
__global__ void deg_kernel(const long long* __restrict__ dst,
                           const float* __restrict__ ew,
                           float* __restrict__ deg, int nE) {
  int e = blockIdx.x * blockDim.x + threadIdx.x;
  if (e < nE) atomicAdd(&deg[(int)dst[e]], ew[e]);
}

// norm[e] = dinv[src] * ew * dinv[dst], dinv = deg>0 ? rsqrt(deg) : 0
__global__ void norm_kernel(const long long* __restrict__ src,
                            const long long* __restrict__ dst,
                            const float* __restrict__ ew,
                            const float* __restrict__ deg,
                            float* __restrict__ nrm, int nE) {
  int e = blockIdx.x * blockDim.x + threadIdx.x;
  if (e < nE) {
    float ds = deg[(int)src[e]];
    float dd = deg[(int)dst[e]];
    float is = ds > 0.f ? rsqrtf(ds) : 0.f;
    float id = dd > 0.f ? rsqrtf(dd) : 0.f;
    nrm[e] = is * ew[e] * id;
  }
}

// ---------------------------------------------------------------------------
// C[rows x 128] = op(A[rows x 128]) @ W[128 x 128]  via V_WMMA_F32_16X16X4_F32
// block = 256 threads = 8 waves; block tile 64 rows x 32 cols; wave tile 16x16
// grid = (rows/64, 128/32). RELU!=0 applies relu to A at load.
// ---------------------------------------------------------------------------
template <int RELU>
__global__ __launch_bounds__(256) void gemm128_wmma(const float* __restrict__ A,
                                                    const float* __restrict__ W,
                                                    float* __restrict__ C) {
  __shared__ float sA[64 * 132];   // 64x128 tile, padded stride 132
  __shared__ float sW[128 * 36];   // 128x32 slab, padded stride 36

  const int tid = threadIdx.x;
  const size_t row0blk = (size_t)blockIdx.x * 64;
  const int colbase = blockIdx.y * 32;

  // stage A tile (2048 float4, 8 per thread), fuse relu
  {
    const float4* A4 = (const float4*)(A + row0blk * F_DIM);
#pragma unroll
    for (int i = 0; i < 8; ++i) {
      int j = tid + 256 * i;
      int r = j >> 5;
      int c = (j & 31) * 4;
      float4 v = A4[j];
      if (RELU) {
        v.x = fmaxf(v.x, 0.f); v.y = fmaxf(v.y, 0.f);
        v.z = fmaxf(v.z, 0.f); v.w = fmaxf(v.w, 0.f);
      }
      *(float4*)(&sA[r * 132 + c]) = v;
    }
  }
  // stage W slab (1024 float4, 4 per thread)
  {
#pragma unroll
    for (int i = 0; i < 4; ++i) {
      int j = tid + 256 * i;
      int r = j >> 3;
      int c = (j & 7) * 4;
      float4 v = *(const float4*)(&W[r * F_DIM + colbase + c]);
      *(float4*)(&sW[r * 36 + c]) = v;
    }
  }
  __syncthreads();

  const int lane = tid & 31;
  const int wv   = tid >> 5;
  const int rt   = wv >> 1;          // row tile 0..3
  const int ct   = wv & 1;           // col tile 0..1
  const int half = lane >> 4;        // 0 / 1
  const int l16  = lane & 15;
  const int mloc = rt * 16 + l16;    // A row (lanes 0-15 hold M=0..15)
  const int cloc = ct * 16 + l16;    // B col (N striped over lanes)

  v8f acc = {0.f, 0.f, 0.f, 0.f, 0.f, 0.f, 0.f, 0.f};
#pragma unroll
  for (int k0 = 0; k0 < 128; k0 += 4) {
    int ka = k0 + half * 2;          // A/B: VGPR pair holds K, K+1; half selects +2
    v2f a, b;
    a.x = sA[mloc * 132 + ka];
    a.y = sA[mloc * 132 + ka + 1];
    b.x = sW[ka * 36 + cloc];
    b.y = sW[(ka + 1) * 36 + cloc];
    acc = __builtin_amdgcn_wmma_f32_16x16x4_f32(
        /*neg_a=*/false, a, /*neg_b=*/false, b,
        /*c_mod=*/(short)0, acc, /*reuse_a=*/false, /*reuse_b=*/false);
  }

  // C/D layout: VGPR i -> M = half*8 + i, N = lane&15
  float* Cp = C + (row0blk + rt * 16 + half * 8) * F_DIM + colbase + ct * 16 + l16;
#pragma unroll
  for (int i = 0; i < 8; ++i) Cp[(size_t)i * F_DIM] = acc[i];
}

// ---------------------------------------------------------------------------
// out[row][f] = bias[f]   (pre-fill accumulation target for scatter)
// ---------------------------------------------------------------------------
__global__ void init_bias_kernel(float* __restrict__ out,
                                 const float* __restrict__ bias, int totalRows) {
  int idx = blockIdx.x * blockDim.x + threadIdx.x;   // one float4 each
  if (idx >= totalRows * 32) return;
  int f4 = idx & 31;
  float4 b = ((const float4*)bias)[f4];
  ((float4*)out)[idx] = b;
}

// ---------------------------------------------------------------------------
// out[t][dst] += norm[e] * H[t][src], one wave per (t, edge), lane = float4
// ---------------------------------------------------------------------------
__global__ void scatter_kernel(const float* __restrict__ H,
                               const long long* __restrict__ src,
                               const long long* __restrict__ dst,
                               const float* __restrict__ nrm,
                               float* __restrict__ out,
                               int nE, int totalWaves) {
  int wid = blockIdx.x * 8 + (threadIdx.x >> 5);
  if (wid >= totalWaves) return;
  int t = wid / nE;
  int e = wid - t * nE;
  float w = nrm[e];
  if (w == 0.f) return;
  int lane = threadIdx.x & 31;
  int s = (int)src[e];
  int d = (int)dst[e];
  size_t sb = ((size_t)t * N_NODES + s) * F_DIM + lane * 4;
  size_t db = ((size_t)t * N_NODES + d) * F_DIM + lane * 4;
  float4 h = *(const float4*)(H + sb);
  float* o = out + db;
  atomicAdd(o + 0, w * h.x);
  atomicAdd(o + 1, w * h.y);
  atomicAdd(o + 2, w * h.z);
  atomicAdd(o + 3, w * h.w);
}

// ---------------------------------------------------------------------------
// Temporal bidirectional chain GCN, analytic norm:
//   d(t) = 1 if t in {0, T-1} else 1/sqrt(2)
//   Z[t] = bias + d(t-1)d(t)*H[t-1] + d(t)d(t+1)*H[t+1]; then relu or tanh
// ---------------------------------------------------------------------------
template <int TANH>
__global__ void temporal_kernel(const float* __restrict__ H,
                                const float* __restrict__ bias,
                                float* __restrict__ out) {
  int idx = blockIdx.x * blockDim.x + threadIdx.x;   // one float4 each
  const int total = T_STEPS * N_NODES * 32;
  if (idx >= total) return;
  int f4 = idx & 31;
  int rn = idx >> 5;
  int t = rn / N_NODES;
  int node = rn - t * N_NODES;

  float4 acc = ((const float4*)bias)[f4];
  const float inv_s2 = 0.70710678118654752f;
  float dt = (t == 0 || t == T_STEPS - 1) ? 1.f : inv_s2;
  if (t > 0) {
    float dp = ((t - 1) == 0 || (t - 1) == T_STEPS - 1) ? 1.f : inv_s2;
    float w = dp * dt;
    float4 h = ((const float4*)H)[((t - 1) * N_NODES + node) * 32 + f4];
    acc.x += w * h.x; acc.y += w * h.y; acc.z += w * h.z; acc.w += w * h.w;
  }
  if (t < T_STEPS - 1) {
    float dn = ((t + 1) == 0 || (t + 1) == T_STEPS - 1) ? 1.f : inv_s2;
    float w = dn * dt;
    float4 h = ((const float4*)H)[((t + 1) * N_NODES + node) * 32 + f4];
    acc.x += w * h.x; acc.y += w * h.y; acc.z += w * h.z; acc.w += w * h.w;
  }
  if (TANH) {
    acc.x = tanhf(acc.x); acc.y = tanhf(acc.y);
    acc.z = tanhf(acc.z); acc.w = tanhf(acc.w);
  } else {
    acc.x = fmaxf(acc.x, 0.f); acc.y = fmaxf(acc.y, 0.f);
    acc.z = fmaxf(acc.z, 0.f); acc.w = fmaxf(acc.w, 0.f);
  }
  ((float4*)out)[idx] = acc;
}

// ---------------------------------------------------------------------------
extern "C" void kernel_launch(void* const* d_in, const int* in_sizes, int n_in,
                              void* d_out, int out_size, void* d_ws, size_t ws_size,
                              hipStream_t stream) {
  (void)n_in; (void)out_size; (void)ws_size;
  const float* x        = (const float*)d_in[0];       // [T,N,128]
  const long long* ei   = (const long long*)d_in[1];   // [2,E] int64
  const float* ew       = (const float*)d_in[2];       // [E]
  const float* W1       = (const float*)d_in[3];
  const float* b1       = (const float*)d_in[4];
  const float* W2       = (const float*)d_in[5];
  const float* b2       = (const float*)d_in[6];
  const float* Wt1      = (const float*)d_in[7];
  const float* bt1      = (const float*)d_in[8];
  const float* Wt2      = (const float*)d_in[9];
  const float* bt2      = (const float*)d_in[10];

  const int nE = in_sizes[2];
  const long long* srcI = ei;
  const long long* dstI = ei + nE;

  const int rows = T_STEPS * N_NODES;                  // 120000, %64 == 0

  // workspace layout: deg | norm | bufA | bufB  (256B aligned)
  char* ws = (char*)d_ws;
  size_t off = 0;
  auto take = [&](size_t bytes) -> char* {
    char* p = ws + off;
    off = (off + bytes + 255) & ~(size_t)255;
    return p;
  };
  float* deg  = (float*)take((size_t)N_NODES * sizeof(float));
  float* nrm  = (float*)take((size_t)nE * sizeof(float));
  float* bufA = (float*)take((size_t)rows * F_DIM * sizeof(float));
  float* bufB = (float*)take((size_t)rows * F_DIM * sizeof(float));

  // edge normalization
  hipMemsetAsync(deg, 0, (size_t)N_NODES * sizeof(float), stream);
  deg_kernel<<<(nE + 255) / 256, 256, 0, stream>>>(dstI, ew, deg, nE);
  norm_kernel<<<(nE + 255) / 256, 256, 0, stream>>>(srcI, dstI, ew, deg, nrm, nE);

  dim3 ggrid(rows / 64, F_DIM / 32);                   // 1875 x 4
  const int elemBlocks = (rows * 32 + 255) / 256;      // 15000
  const int totalWaves = T_STEPS * nE;                 // 1.92M
  const int scatBlocks = (totalWaves + 7) / 8;

  // spatial GCN layer 1: relu(scatter(norm * (x@W1)[src]) + b1)
  gemm128_wmma<0><<<ggrid, 256, 0, stream>>>(x, W1, bufA);
  init_bias_kernel<<<elemBlocks, 256, 0, stream>>>(bufB, b1, rows);
  scatter_kernel<<<scatBlocks, 256, 0, stream>>>(bufA, srcI, dstI, nrm, bufB, nE, totalWaves);

  // spatial GCN layer 2 (relu fused into A load): scatter(norm * (relu(h)@W2)[src]) + b2
  gemm128_wmma<1><<<ggrid, 256, 0, stream>>>(bufB, W2, bufA);
  init_bias_kernel<<<elemBlocks, 256, 0, stream>>>(bufB, b2, rows);
  scatter_kernel<<<scatBlocks, 256, 0, stream>>>(bufA, srcI, dstI, nrm, bufB, nE, totalWaves);

  // temporal GCN layer 1: relu(combine(xf@Wt1) + bt1)
  gemm128_wmma<0><<<ggrid, 256, 0, stream>>>(bufB, Wt1, bufA);
  temporal_kernel<0><<<elemBlocks, 256, 0, stream>>>(bufA, bt1, bufB);

  // temporal GCN layer 2 + tanh -> output
  gemm128_wmma<0><<<ggrid, 256, 0, stream>>>(bufB, Wt2, bufA);
  temporal_kernel<1><<<elemBlocks, 256, 0, stream>>>(bufA, bt2, (float*)d_out);
}